// TransitionDown_7645041787058
// MI455X (gfx1250) — compile-verified
//
#include <hip/hip_runtime.h>
#include <hip/hip_bf16.h>
#include <math.h>

typedef float v2f __attribute__((ext_vector_type(2)));
typedef float v4f __attribute__((ext_vector_type(4)));
typedef float v8f __attribute__((ext_vector_type(8)));

#define BATCH 16
#define NPTS  4096
#define CIN   256
#define COUT  512
#define KNBR  16
#define NOUT  1024
#define MROWS (BATCH * NPTS)       /* 65536 */
#define BN_EPS 1e-5f
#define XOUT_ELEMS ((size_t)BATCH * NOUT * COUT)  /* 8388608 */

// ---------------------------------------------------------------------------
// Kernel 1: h = relu( (x @ W^T + b - mean) * gamma*rsqrt(var+eps) + beta )
// WMMA f32 16x16x4 path. One wave computes a 16(M) x 64(N) strip:
// 4 accumulators (v8f each), A fragment reused across the 4 N-tiles.
// grid = (MROWS/64, COUT/64), block = 128 threads (4 waves).
// ---------------------------------------------------------------------------
__global__ __launch_bounds__(128) void gemm_bn_relu_kernel(
    const float* __restrict__ x, const float* __restrict__ W,
    const float* __restrict__ bias, const float* __restrict__ gamma,
    const float* __restrict__ beta, const float* __restrict__ rmean,
    const float* __restrict__ rvar, float* __restrict__ h)
{
    const int wave = threadIdx.x >> 5;
    const int lane = threadIdx.x & 31;
    const int m0 = (blockIdx.x * 4 + wave) * 16;
    const int n0 = blockIdx.y * 64;

    // A fragment (f32 16x4, 2 VGPRs/lane): lanes 0-15 -> K=0,1 ; lanes 16-31 -> K=2,3
    const int rowA = m0 + (lane & 15);
    const int kOff = (lane >> 4) << 1;          // 0 or 2
    const float* aPtr = x + (size_t)rowA * CIN + kOff;

    // B fragment (f32 4x16): lane -> column N, VGPR/lane-half -> K (mirrors A)
    const int colL = lane & 15;
    const float* bPtr0 = W + (size_t)(n0 +  0 + colL) * CIN + kOff;
    const float* bPtr1 = W + (size_t)(n0 + 16 + colL) * CIN + kOff;
    const float* bPtr2 = W + (size_t)(n0 + 32 + colL) * CIN + kOff;
    const float* bPtr3 = W + (size_t)(n0 + 48 + colL) * CIN + kOff;

    v8f acc[4];
    #pragma unroll
    for (int nt = 0; nt < 4; ++nt) acc[nt] = {};

    #pragma unroll 4
    for (int c = 0; c < CIN; c += 4) {
        v2f a  = *(const v2f*)(aPtr + c);
        v2f b0 = *(const v2f*)(bPtr0 + c);
        v2f b1 = *(const v2f*)(bPtr1 + c);
        v2f b2 = *(const v2f*)(bPtr2 + c);
        v2f b3 = *(const v2f*)(bPtr3 + c);
        acc[0] = __builtin_amdgcn_wmma_f32_16x16x4_f32(false, a, false, b0,
                                                       (short)0, acc[0], false, false);
        acc[1] = __builtin_amdgcn_wmma_f32_16x16x4_f32(false, a, false, b1,
                                                       (short)0, acc[1], false, false);
        acc[2] = __builtin_amdgcn_wmma_f32_16x16x4_f32(false, a, false, b2,
                                                       (short)0, acc[2], false, false);
        acc[3] = __builtin_amdgcn_wmma_f32_16x16x4_f32(false, a, false, b3,
                                                       (short)0, acc[3], false, false);
    }

    // Epilogue. C/D layout: VGPR j holds rows j (lanes 0-15) and j+8 (lanes 16-31),
    // column = lane & 15 within the tile (constant across j).
    const int rOff = (lane >> 4) << 3;          // 0 or 8
    #pragma unroll
    for (int nt = 0; nt < 4; ++nt) {
        const int o = n0 + nt * 16 + colL;
        const float sc = gamma[o] * rsqrtf(rvar[o] + BN_EPS);
        const float sh = (bias[o] - rmean[o]) * sc + beta[o];
        #pragma unroll
        for (int j = 0; j < 8; ++j) {
            float v = fmaf(acc[nt][j], sc, sh);
            v = fmaxf(v, 0.0f);
            h[(size_t)(m0 + j + rOff) * COUT + o] = v;
        }
    }
}

// ---------------------------------------------------------------------------
// Kernel 2: farthest point sampling. One block (1024 thr = 32 waves) per batch.
// Each thread owns 4 points in registers. Argmax reduction matches jnp.argmax
// first-occurrence tie-break via lexicographic (value desc, index asc) merge.
// Writes GLOBAL flat indices fid[b*NOUT + i] = b*NPTS + local.
// ---------------------------------------------------------------------------
__global__ __launch_bounds__(1024) void fps_kernel(const float* __restrict__ p,
                                                   int* __restrict__ fid)
{
    const int b = blockIdx.x;
    const float* pb = p + (size_t)b * NPTS * 3;
    const int t = threadIdx.x;

    float px[4], py[4], pz[4], d[4];
    const float x0 = pb[0], y0 = pb[1], z0 = pb[2];
    #pragma unroll
    for (int s = 0; s < 4; ++s) {
        const int idx = t + s * 1024;
        px[s] = pb[idx * 3 + 0];
        py[s] = pb[idx * 3 + 1];
        pz[s] = pb[idx * 3 + 2];
        const float dx = px[s] - x0, dy = py[s] - y0, dz = pz[s] - z0;
        d[s] = dx * dx + dy * dy + dz * dz;
    }

    __shared__ float sval[32];
    __shared__ int   sidx[32];
    __shared__ int   snxt;
    __shared__ float scoord[3];

    if (t == 0) fid[b * NOUT + 0] = b * NPTS;   // idx[0] = 0

    const int wid = t >> 5, lane = t & 31;

    for (int i = 1; i < NOUT; ++i) {
        // local argmax (first index wins on ties: ascending s, strict >)
        float bv = d[0]; int bi = t;
        #pragma unroll
        for (int s = 1; s < 4; ++s) {
            if (d[s] > bv) { bv = d[s]; bi = t + s * 1024; }
        }
        // intra-wave butterfly reduce (no barrier needed, wave32)
        #pragma unroll
        for (int off = 16; off >= 1; off >>= 1) {
            const float ov = __shfl_xor(bv, off, 32);
            const int   oi = __shfl_xor(bi, off, 32);
            if (ov > bv || (ov == bv && oi < bi)) { bv = ov; bi = oi; }
        }
        if (lane == 0) { sval[wid] = bv; sidx[wid] = bi; }
        __syncthreads();
        if (wid == 0) {
            float v = sval[lane]; int ix = sidx[lane];
            #pragma unroll
            for (int off = 16; off >= 1; off >>= 1) {
                const float ov = __shfl_xor(v, off, 32);
                const int   oi = __shfl_xor(ix, off, 32);
                if (ov > v || (ov == v && oi < ix)) { v = ov; ix = oi; }
            }
            if (lane == 0) {
                snxt = ix;
                scoord[0] = pb[ix * 3 + 0];
                scoord[1] = pb[ix * 3 + 1];
                scoord[2] = pb[ix * 3 + 2];
                fid[b * NOUT + i] = b * NPTS + ix;
            }
        }
        __syncthreads();
        const float nx = scoord[0], ny = scoord[1], nz = scoord[2];
        #pragma unroll
        for (int s = 0; s < 4; ++s) {
            const float dx = px[s] - nx, dy = py[s] - ny, dz = pz[s] - nz;
            const float dd = dx * dx + dy * dy + dz * dz;
            d[s] = fminf(d[s], dd);
        }
    }
}

// ---------------------------------------------------------------------------
// Kernel 3: per sampled point, gather K=16 neighbor rows of h, max over K.
// One block (128 thr) per sample; thread t handles channels [4t, 4t+4) with
// 128-bit loads (h is L2-resident: 128 MB < 192 MB L2). Also emits p_out.
// ---------------------------------------------------------------------------
__global__ __launch_bounds__(128) void gather_maxpool_kernel(
    const float* __restrict__ h, const float* __restrict__ p,
    const int* __restrict__ fid, const int* __restrict__ sid_euc,
    float* __restrict__ out)
{
    const int s = blockIdx.x;          // 0 .. BATCH*NOUT-1
    const int t = threadIdx.x;         // 0 .. 127

    __shared__ int ssid[KNBR];
    const int f = fid[s];              // global row in [0, B*N)
    if (t < KNBR) ssid[t] = sid_euc[(size_t)f * KNBR + t];
    if (t < 3)    out[XOUT_ELEMS + (size_t)s * 3 + t] = p[(size_t)f * 3 + t];
    __syncthreads();

    const int c = t * 4;
    v4f acc = { -INFINITY, -INFINITY, -INFINITY, -INFINITY };
    #pragma unroll
    for (int k = 0; k < KNBR; ++k) {
        const int r = ssid[k];
        if (k + 1 < KNBR)   // hint next random row toward L2/L0
            __builtin_prefetch(h + (size_t)ssid[k + 1] * COUT + c, 0, 0);
        const v4f v = *(const v4f*)(h + (size_t)r * COUT + c);
        acc.x = fmaxf(acc.x, v.x);
        acc.y = fmaxf(acc.y, v.y);
        acc.z = fmaxf(acc.z, v.z);
        acc.w = fmaxf(acc.w, v.w);
    }
    *(v4f*)(out + (size_t)s * COUT + c) = acc;
}

// ---------------------------------------------------------------------------
extern "C" void kernel_launch(void* const* d_in, const int* in_sizes, int n_in,
                              void* d_out, int out_size, void* d_ws, size_t ws_size,
                              hipStream_t stream)
{
    (void)in_sizes; (void)n_in; (void)out_size; (void)ws_size;
    const float* x     = (const float*)d_in[0];  // [B,N,CIN]
    const float* p     = (const float*)d_in[1];  // [B,N,3]
    const int*   sid   = (const int*)  d_in[2];  // [B*N,K]
    /* d_in[3] = tid_euc (unused by reference path) */
    const float* W     = (const float*)d_in[4];  // [COUT,CIN]
    const float* bias  = (const float*)d_in[5];
    const float* gamma = (const float*)d_in[6];
    const float* beta  = (const float*)d_in[7];
    const float* rmean = (const float*)d_in[8];
    const float* rvar  = (const float*)d_in[9];
    float* out = (float*)d_out;

    float* h  = (float*)d_ws;                                   // 65536*512 f32 = 128 MB
    int* fid  = (int*)((char*)d_ws + (size_t)MROWS * COUT * 4); // + 64 KB

    dim3 ggrid(MROWS / 64, COUT / 64);                          // (1024, 8)
    gemm_bn_relu_kernel<<<ggrid, 128, 0, stream>>>(x, W, bias, gamma, beta,
                                                   rmean, rvar, h);
    fps_kernel<<<BATCH, 1024, 0, stream>>>(p, fid);
    gather_maxpool_kernel<<<BATCH * NOUT, 128, 0, stream>>>(h, p, fid, sid, out);
}